// DynamicBlock_80315888435314
// MI455X (gfx1250) — compile-verified
//
#include <hip/hip_runtime.h>
#include <math.h>

// Problem constants (match reference).
#define B_    4
#define T_    2048
#define D_    1024
#define H_    16
#define KV_   8
#define HD_   64
#define DFF_  4096
#define KSEL_ 512

typedef __attribute__((ext_vector_type(2))) float v2f;
typedef __attribute__((ext_vector_type(4))) float v4f;
typedef __attribute__((ext_vector_type(8))) float v8f;
typedef __attribute__((ext_vector_type(4))) int   v4i;

__device__ __forceinline__ v8f wmma4(v2f a, v2f b, v8f c) {
  // D = A(16x4) * B(4x16) + C, fp32 in / fp32 accumulate.
  return __builtin_amdgcn_wmma_f32_16x16x4_f32(false, a, false, b, (short)0, c, false, false);
}

// ---- gfx1250 async global->LDS copy (ASYNCcnt) with portable fallback ----
#if __has_builtin(__builtin_amdgcn_global_load_async_to_lds_b128) && \
    __has_builtin(__builtin_amdgcn_s_wait_asynccnt)
#define USE_ASYNC_LDS 1
#else
#define USE_ASYNC_LDS 0
#endif

#if USE_ASYNC_LDS
typedef __attribute__((address_space(1))) v4i gv4i_t;  // global 128-bit vector
typedef __attribute__((address_space(3))) v4i lv4i_t;  // LDS 128-bit vector
#endif

__device__ __forceinline__ void copy16_g2l(const float* g, float* l) {
#if USE_ASYNC_LDS
  __builtin_amdgcn_global_load_async_to_lds_b128((gv4i_t*)g, (lv4i_t*)l, 0, 0);
#else
  *(v4f*)l = *(const v4f*)g;
#endif
}

__device__ __forceinline__ void async_wait_4() {
#if USE_ASYNC_LDS
  __builtin_amdgcn_s_wait_asynccnt(4);
#endif
}
__device__ __forceinline__ void async_wait_0() {
#if USE_ASYNC_LDS
  __builtin_amdgcn_s_wait_asynccnt(0);
#endif
}

// ---------------------------------------------------------------- utilities

__global__ __launch_bounds__(256) void build_rowmap_kernel(
    const int* __restrict__ tok, const int* __restrict__ bat,
    int* __restrict__ map, int n) {
  int i = blockIdx.x * 256 + threadIdx.x;
  if (i < n) map[i] = bat[i] * T_ + tok[i];
}

__global__ __launch_bounds__(256) void rmsnorm_kernel(
    const float* __restrict__ x, const float* __restrict__ w,
    float* __restrict__ y) {
  const int row = blockIdx.x;
  const float* xr = x + (long long)row * D_;
  float* yr = y + (long long)row * D_;
  float s = 0.f;
  for (int c = threadIdx.x; c < D_; c += 256) { float v = xr[c]; s += v * v; }
  s += __shfl_xor(s, 16, 32);
  s += __shfl_xor(s, 8, 32);
  s += __shfl_xor(s, 4, 32);
  s += __shfl_xor(s, 2, 32);
  s += __shfl_xor(s, 1, 32);
  __shared__ float red[8];
  __shared__ float invs;
  const int wave = threadIdx.x >> 5;
  if ((threadIdx.x & 31) == 0) red[wave] = s;
  __syncthreads();
  if (threadIdx.x == 0) {
    float t = 0.f;
    for (int i = 0; i < 8; ++i) t += red[i];
    invs = rsqrtf(t / (float)D_ + 1e-6f);
  }
  __syncthreads();
  const float inv = invs;
  for (int c = threadIdx.x; c < D_; c += 256) yr[c] = xr[c] * inv * w[c];
}

// RoPE on gathered Q rows: q[m, h, :], pos = tok[m], batch = bat[m].
__global__ __launch_bounds__(256) void rope_q_kernel(
    float* __restrict__ q, const int* __restrict__ tok, const int* __restrict__ bat,
    const float* __restrict__ cs, const float* __restrict__ sn) {
  int gid = blockIdx.x * 256 + threadIdx.x;
  int d = gid & 31;
  int h = (gid >> 5) & (H_ - 1);
  int m = gid >> 9;
  if (m >= B_ * KSEL_) return;
  int b = bat[m], t = tok[m];
  const float* cp = cs + ((long long)b * T_ + t) * HD_;
  const float* sp = sn + ((long long)b * T_ + t) * HD_;
  float* p = q + (long long)m * (H_ * HD_) + h * HD_;
  float x1 = p[d], x2 = p[d + 32];
  p[d]      = x1 * cp[d]      - x2 * sp[d];
  p[d + 32] = x2 * cp[d + 32] + x1 * sp[d + 32];
}

// RoPE on all K rows: row = b*T + pos.
__global__ __launch_bounds__(256) void rope_k_kernel(
    float* __restrict__ k,
    const float* __restrict__ cs, const float* __restrict__ sn) {
  int gid = blockIdx.x * 256 + threadIdx.x;
  int d = gid & 31;
  int kh = (gid >> 5) & (KV_ - 1);
  int row = gid >> 8;
  if (row >= B_ * T_) return;
  const float* cp = cs + (long long)row * HD_;
  const float* sp = sn + (long long)row * HD_;
  float* p = k + (long long)row * (KV_ * HD_) + kh * HD_;
  float x1 = p[d], x2 = p[d + 32];
  p[d]      = x1 * cp[d]      - x2 * sp[d];
  p[d + 32] = x2 * cp[d + 32] + x1 * sp[d + 32];
}

__global__ __launch_bounds__(256) void mul_kernel(
    float* __restrict__ a, const float* __restrict__ b, int n) {
  int i = blockIdx.x * 256 + threadIdx.x;
  if (i < n) a[i] *= b[i];
}

// ---------------------------------------------------------------- WMMA GEMM
// C[M,N] = epilogue(A[M,K] @ W[N,K]^T + bias)
// Block = 128 threads (4 waves). Block tile = 32(M) x 256(N); each wave owns
// 32x64 (two 16-row WMMA subtiles sharing the B fragments).
// A k-chunks (32 x 64) are staged through LDS with double-buffered async DMA.
// grid.x = M/32, grid.y = N/256.  K % 64 == 0.

#define KCHUNK 64
#define APAD   68  // padded LDS row stride (floats): 16B-aligned, bank-conflict-free

enum { EP_NONE = 0, EP_SILU = 1, EP_RES = 2, EP_FINAL = 3 };

__device__ __forceinline__ void stage_chunk(const float* __restrict__ A, int K,
                                            int kbase, float* __restrict__ buf,
                                            const int* __restrict__ rowbase, int tid) {
  // 32 rows x 64 floats = 512 x 16B segments, 128 threads x 4 segments.
#pragma unroll
  for (int i = 0; i < 4; ++i) {
    const int seg = i * 128 + tid;
    const int row = seg >> 4;
    const int s16 = seg & 15;
    const float* g = A + (long long)rowbase[row] * K + kbase + s16 * 4;
    copy16_g2l(g, buf + row * APAD + s16 * 4);
  }
}

template <int EPI>
__global__ __launch_bounds__(128) void gemm_wmma_kernel(
    const float* __restrict__ A, const float* __restrict__ W,
    const float* __restrict__ bias, float* __restrict__ C,
    int N, int K,
    const int* __restrict__ agather,  // optional: A row gather (Q projection)
    const int* __restrict__ omap,     // EP_RES / EP_FINAL: selected-row -> b*T+tok
    const float* __restrict__ R,      // EP_RES: residual (hidden); EP_FINAL: hidden
    const float* __restrict__ Hb,     // EP_FINAL: h buffer (m-indexed)
    const float* __restrict__ gsc) {  // EP_FINAL: gating scores (m-indexed)
  __shared__ __align__(16) float abuf[2][32 * APAD];
  __shared__ int rowbase[32];

  const int tid  = threadIdx.x;
  const int lane = tid & 31;
  const int wave = tid >> 5;
  const int half = lane >> 4;
  const int mlo  = lane & 15;
  const int m0 = blockIdx.x * 32;
  const int n0 = blockIdx.y * 256 + wave * 64;

  if (tid < 32) {
    const int am = m0 + tid;
    rowbase[tid] = agather ? agather[am] : am;
  }
  __syncthreads();

  const float* Wr[4];
#pragma unroll
  for (int nt = 0; nt < 4; ++nt)
    Wr[nt] = W + (long long)(n0 + nt * 16 + mlo) * K + 2 * half;

  v8f zero = {};
  v8f acc0[4] = {zero, zero, zero, zero};  // rows m0 .. m0+15
  v8f acc1[4] = {zero, zero, zero, zero};  // rows m0+16 .. m0+31

  const int nch = K / KCHUNK;
  stage_chunk(A, K, 0, abuf[0], rowbase, tid);

  for (int c = 0; c < nch; ++c) {
    const int kbase = c * KCHUNK;
    if (c + 1 < nch) {
      stage_chunk(A, K, kbase + KCHUNK, abuf[(c + 1) & 1], rowbase, tid);
#pragma unroll
      for (int nt = 0; nt < 4; ++nt)  // pull next W chunk toward the WGP
        __builtin_prefetch(Wr[nt] + kbase + KCHUNK, 0, 0);
      async_wait_4();  // chunk c done (in-order), chunk c+1 still in flight
    } else {
      async_wait_0();
    }
    __syncthreads();

    const float* ab  = abuf[c & 1];
    const float* a0p = ab + mlo * APAD + 2 * half;
    const float* a1p = ab + (mlo + 16) * APAD + 2 * half;
#pragma unroll
    for (int k0 = 0; k0 < KCHUNK; k0 += 4) {
      v2f a0 = *(const v2f*)(a0p + k0);
      v2f a1 = *(const v2f*)(a1p + k0);
#pragma unroll
      for (int nt = 0; nt < 4; ++nt) {
        v2f b = *(const v2f*)(Wr[nt] + kbase + k0);
        acc0[nt] = wmma4(a0, b, acc0[nt]);
        acc1[nt] = wmma4(a1, b, acc1[nt]);
      }
    }
    __syncthreads();  // before buf[c&1] is overwritten by chunk c+2's DMA
  }

#pragma unroll
  for (int sub = 0; sub < 2; ++sub) {
    const v8f* accs = (sub == 0) ? acc0 : acc1;
#pragma unroll
    for (int nt = 0; nt < 4; ++nt) {
      const int n = n0 + nt * 16 + mlo;
      const float bv = bias ? bias[n] : 0.f;
#pragma unroll
      for (int j = 0; j < 8; ++j) {
        const int m = m0 + sub * 16 + j + 8 * half;  // D-layout row
        float v = accs[nt][j] + bv;
        if (EPI == EP_SILU) v = v / (1.f + __expf(-v));
        if (EPI == EP_NONE || EPI == EP_SILU) {
          C[(long long)m * N + n] = v;
        } else if (EPI == EP_RES) {
          const int g = omap[m];
          C[(long long)m * N + n] = v + R[(long long)g * N + n];
        } else {  // EP_FINAL: out[g] = hid + ((h + mlp) - hid) * gate
          const int g = omap[m];
          const float hid = R[(long long)g * N + n];
          const float hv  = Hb[(long long)m * N + n];
          C[(long long)g * N + n] = hid + (hv + v - hid) * gsc[m];
        }
      }
    }
  }
}

// ------------------------------------------------------- flash attention
// One wave (32 threads) per (b, h, tile of 16 selected tokens).
// S = Q*K^T and P*V run on V_WMMA_F32_16X16X4_F32; online softmax between.
// S uses 4 independent partial accumulators to break the 16-deep WMMA chain.

__global__ __launch_bounds__(32) void attn_kernel(
    const float* __restrict__ Q,    // [B*KSEL, H*HD] (roped, gathered)
    const float* __restrict__ Kb,   // [B*T, KV*HD]   (roped)
    const float* __restrict__ Vb,   // [B*T, KV*HD]
    const int* __restrict__ tok,    // [B*KSEL] sorted per batch
    float* __restrict__ ctx) {      // [B*KSEL, H*HD]
  const int lane = threadIdx.x;
  const int half = lane >> 4;
  const int mlo  = lane & 15;
  const int tile = blockIdx.x;
  const int h    = blockIdx.y;
  const int b    = blockIdx.z;
  const int kvh  = h >> 1;  // H/KV == 2
  const int selbase = b * KSEL_ + tile * 16;

  __shared__ float pb[256];  // 16x16 P tile bounce for D-layout -> A-frag

  int tk[8];
#pragma unroll
  for (int r = 0; r < 8; ++r) tk[r] = tok[selbase + r + 8 * half];
  const int tmax = tok[selbase + 15];  // sorted ascending within batch

  v2f qa[16];
  const float* qr = Q + (long long)(selbase + mlo) * (H_ * HD_) + h * HD_ + 2 * half;
#pragma unroll
  for (int c = 0; c < 16; ++c) qa[c] = *(const v2f*)(qr + c * 4);

  v8f zero = {};
  v8f acc[4] = {zero, zero, zero, zero};
  float mrow[8], lrow[8];
#pragma unroll
  for (int r = 0; r < 8; ++r) { mrow[r] = -3.0e38f; lrow[r] = 0.f; }

  for (int t0 = 0; t0 <= tmax; t0 += 16) {
    // ---- S = Q * K^T (16x16), K dim = 64; 4 independent WMMA chains
    v8f sp0 = zero, sp1 = zero, sp2 = zero, sp3 = zero;
    const float* kr = Kb + ((long long)(b * T_ + t0 + mlo) * KV_ + kvh) * HD_ + 2 * half;
#pragma unroll
    for (int c = 0; c < 4; ++c) {
      v2f k0 = *(const v2f*)(kr + (4 * c + 0) * 4);
      v2f k1 = *(const v2f*)(kr + (4 * c + 1) * 4);
      v2f k2 = *(const v2f*)(kr + (4 * c + 2) * 4);
      v2f k3 = *(const v2f*)(kr + (4 * c + 3) * 4);
      sp0 = wmma4(qa[4 * c + 0], k0, sp0);
      sp1 = wmma4(qa[4 * c + 1], k1, sp1);
      sp2 = wmma4(qa[4 * c + 2], k2, sp2);
      sp3 = wmma4(qa[4 * c + 3], k3, sp3);
    }
    v8f s = (sp0 + sp1) + (sp2 + sp3);

    // ---- masked online softmax (row = r + 8*half, col = mlo)
    float alpha[8];
#pragma unroll
    for (int r = 0; r < 8; ++r) {
      float sv = s[r] * 0.125f;                       // 1/sqrt(64)
      sv = (tk[r] >= t0 + mlo) ? sv : -3.0e38f;       // causal mask
      float rm = sv;
      rm = fmaxf(rm, __shfl_xor(rm, 1, 32));
      rm = fmaxf(rm, __shfl_xor(rm, 2, 32));
      rm = fmaxf(rm, __shfl_xor(rm, 4, 32));
      rm = fmaxf(rm, __shfl_xor(rm, 8, 32));
      const float mnew = fmaxf(mrow[r], rm);
      const float p  = __expf(sv - mnew);
      const float al = __expf(mrow[r] - mnew);
      mrow[r] = mnew;
      float ps = p;
      ps += __shfl_xor(ps, 1, 32);
      ps += __shfl_xor(ps, 2, 32);
      ps += __shfl_xor(ps, 4, 32);
      ps += __shfl_xor(ps, 8, 32);
      lrow[r] = lrow[r] * al + ps;
      alpha[r] = al;
      pb[(r + 8 * half) * 16 + mlo] = p;  // P row-major [m][t-local]
    }
#pragma unroll
    for (int r = 0; r < 8; ++r) {
      acc[0][r] *= alpha[r];
      acc[1][r] *= alpha[r];
      acc[2][r] *= alpha[r];
      acc[3][r] *= alpha[r];
    }
    __syncthreads();

    // ---- ctx += P(16x16) * V(16x64); 4 independent chains (one per n-tile)
#pragma unroll
    for (int kc = 0; kc < 4; ++kc) {
      v2f pa = *(const v2f*)(&pb[mlo * 16 + kc * 4 + 2 * half]);  // P as A-frag
      const int trow = t0 + kc * 4 + 2 * half;
      const float* v0 = Vb + ((long long)(b * T_ + trow) * KV_ + kvh) * HD_;
      const float* v1 = Vb + ((long long)(b * T_ + trow + 1) * KV_ + kvh) * HD_;
#pragma unroll
      for (int nt = 0; nt < 4; ++nt) {
        v2f bv;
        bv.x = v0[nt * 16 + mlo];
        bv.y = v1[nt * 16 + mlo];
        acc[nt] = wmma4(pa, bv, acc[nt]);
      }
    }
    __syncthreads();
  }

#pragma unroll
  for (int nt = 0; nt < 4; ++nt) {
#pragma unroll
    for (int r = 0; r < 8; ++r) {
      const int m = r + 8 * half;
      ctx[(long long)(selbase + m) * (H_ * HD_) + h * HD_ + nt * 16 + mlo] =
          acc[nt][r] / lrow[r];
    }
  }
}

// ---------------------------------------------------------------- launcher

extern "C" void kernel_launch(void* const* d_in, const int* in_sizes, int n_in,
                              void* d_out, int out_size, void* d_ws, size_t ws_size,
                              hipStream_t stream) {
  (void)in_sizes; (void)n_in; (void)out_size; (void)ws_size;
  const float* hidden = (const float*)d_in[0];
  const int*   tok    = (const int*)d_in[1];
  const int*   bat    = (const int*)d_in[2];
  const float* gsc    = (const float*)d_in[3];
  const float* cosp   = (const float*)d_in[4];
  const float* sinp   = (const float*)d_in[5];
  const float* ln1    = (const float*)d_in[6];
  const float* ln2    = (const float*)d_in[7];
  const float* q_w    = (const float*)d_in[8];
  const float* q_b    = (const float*)d_in[9];
  const float* k_w    = (const float*)d_in[10];
  const float* k_b    = (const float*)d_in[11];
  const float* v_w    = (const float*)d_in[12];
  const float* v_b    = (const float*)d_in[13];
  const float* o_w    = (const float*)d_in[14];
  const float* gate_w = (const float*)d_in[15];
  const float* up_w   = (const float*)d_in[16];
  const float* down_w = (const float*)d_in[17];
  float* out = (float*)d_out;

  const int NTOK = B_ * T_;      // 8192
  const int NSEL = B_ * KSEL_;   // 2048

  float* ws = (float*)d_ws;
  size_t off = 0;
  float* normed = ws + off; off += (size_t)NTOK * D_;        // 8192x1024
  float* Kbuf   = ws + off; off += (size_t)NTOK * KV_ * HD_; // 8192x512
  float* Vbuf   = ws + off; off += (size_t)NTOK * KV_ * HD_;
  float* Qsel   = ws + off; off += (size_t)NSEL * H_ * HD_;  // 2048x1024
  float* ctx    = ws + off; off += (size_t)NSEL * H_ * HD_;
  float* Hbuf   = ws + off; off += (size_t)NSEL * D_;
  float* n2     = ws + off; off += (size_t)NSEL * D_;
  float* gact   = ws + off; off += (size_t)NSEL * DFF_;      // 2048x4096
  float* ubuf   = ws + off; off += (size_t)NSEL * DFF_;
  int*   selmap = (int*)(ws + off);

  // 0) out starts as a copy of hidden; selected rows get overwritten later.
  (void)hipMemcpyAsync(out, hidden, (size_t)NTOK * D_ * sizeof(float),
                       hipMemcpyDeviceToDevice, stream);

  // 1) rowmap: selected row -> b*T + tok
  build_rowmap_kernel<<<dim3((NSEL + 255) / 256), 256, 0, stream>>>(tok, bat, selmap, NSEL);

  // 2) rmsnorm over all tokens
  rmsnorm_kernel<<<dim3(NTOK), 256, 0, stream>>>(hidden, ln1, normed);

  // 3) K, V projections over all tokens; Q over gathered selected rows
  gemm_wmma_kernel<EP_NONE><<<dim3(NTOK / 32, (KV_ * HD_) / 256), 128, 0, stream>>>(
      normed, k_w, k_b, Kbuf, KV_ * HD_, D_, nullptr, nullptr, nullptr, nullptr, nullptr);
  gemm_wmma_kernel<EP_NONE><<<dim3(NTOK / 32, (KV_ * HD_) / 256), 128, 0, stream>>>(
      normed, v_w, v_b, Vbuf, KV_ * HD_, D_, nullptr, nullptr, nullptr, nullptr, nullptr);
  gemm_wmma_kernel<EP_NONE><<<dim3(NSEL / 32, (H_ * HD_) / 256), 128, 0, stream>>>(
      normed, q_w, q_b, Qsel, H_ * HD_, D_, selmap, nullptr, nullptr, nullptr, nullptr);

  // 4) RoPE
  rope_k_kernel<<<dim3((NTOK * KV_ * 32) / 256), 256, 0, stream>>>(Kbuf, cosp, sinp);
  rope_q_kernel<<<dim3((NSEL * H_ * 32) / 256), 256, 0, stream>>>(Qsel, tok, bat, cosp, sinp);

  // 5) flash attention -> ctx
  attn_kernel<<<dim3(KSEL_ / 16, H_, B_), 32, 0, stream>>>(Qsel, Kbuf, Vbuf, tok, ctx);

  // 6) O projection + gathered residual -> h
  gemm_wmma_kernel<EP_RES><<<dim3(NSEL / 32, D_ / 256), 128, 0, stream>>>(
      ctx, o_w, nullptr, Hbuf, D_, H_ * HD_, nullptr, selmap, hidden, nullptr, nullptr);

  // 7) rmsnorm2
  rmsnorm_kernel<<<dim3(NSEL), 256, 0, stream>>>(Hbuf, ln2, n2);

  // 8) MLP: silu(n2@gate^T) * (n2@up^T) @ down^T, fused gating scatter into out
  gemm_wmma_kernel<EP_SILU><<<dim3(NSEL / 32, DFF_ / 256), 128, 0, stream>>>(
      n2, gate_w, nullptr, gact, DFF_, D_, nullptr, nullptr, nullptr, nullptr, nullptr);
  gemm_wmma_kernel<EP_NONE><<<dim3(NSEL / 32, DFF_ / 256), 128, 0, stream>>>(
      n2, up_w, nullptr, ubuf, DFF_, D_, nullptr, nullptr, nullptr, nullptr, nullptr);
  mul_kernel<<<dim3((NSEL * DFF_) / 256), 256, 0, stream>>>(gact, ubuf, NSEL * DFF_);
  gemm_wmma_kernel<EP_FINAL><<<dim3(NSEL / 32, D_ / 256), 128, 0, stream>>>(
      gact, down_w, nullptr, out, D_, DFF_, nullptr, selmap, hidden, Hbuf, gsc);
}